// ProposalLayer_61409442398729
// MI455X (gfx1250) — compile-verified
//
#include <hip/hip_runtime.h>
#include <hip/hip_bf16.h>
#include <math.h>

typedef __attribute__((ext_vector_type(2))) float v2f;
typedef __attribute__((ext_vector_type(8))) float v8f;

#define FHH 50
#define FWW 50
#define NA 9
#define NANCH (FHH*FWW*NA)     // 22500
#define NSORT 32768            // next pow2 >= 22500
#define PRE_NMS 6000
#define PRE_PAD 6016           // 188 * 32
#define NWORDS 188
#define POST_NMS 300
#define NBATCH 4
#define NMS_THR 0.7f

// ---------------------------------------------------------------- anchors
__device__ __forceinline__ float4 anchor_at(int i) {
  // py-faster-rcnn template, base=16: ws/hs rounded per ratio {0.5,1,2}
  const float wsv[3] = {23.f, 16.f, 11.f};
  const float hsv[3] = {12.f, 16.f, 22.f};
  const float scv[3] = {8.f, 16.f, 32.f};
  int a = i % NA; int cell = i / NA;
  int ix = cell % FWW; int iy = cell / FWW;
  int r = a / 3, s = a % 3;
  float W = wsv[r] * scv[s], H = hsv[r] * scv[s];
  float cx = 7.5f + (float)ix * 16.f;
  float cy = 7.5f + (float)iy * 16.f;
  return make_float4(cx - 0.5f*(W-1.f), cy - 0.5f*(H-1.f),
                     cx + 0.5f*(W-1.f), cy + 0.5f*(H-1.f));
}

__global__ void anchors_kernel(float* __restrict__ out) {
  int i = blockIdx.x * blockDim.x + threadIdx.x;
  if (i >= NANCH) return;
  float4 a = anchor_at(i);
  // anchors live at floats [6000, 96000) of d_out
  float* dst = out + 6000 + i * 4;
  dst[0] = a.x; dst[1] = a.y; dst[2] = a.z; dst[3] = a.w;
}

// ------------------------------------------------- decode + clip + filter
__global__ void decode_kernel(const float* __restrict__ deltas,
                              const float* __restrict__ scores,
                              const int* __restrict__ ph, const int* __restrict__ pw,
                              float* __restrict__ boxes,   // [B][NANCH][4]
                              float* __restrict__ keys,    // [B][NSORT]
                              int*   __restrict__ idxs) {  // [B][NSORT]
  int g = blockIdx.x * blockDim.x + threadIdx.x;
  int b = g >> 15;            // /NSORT
  int n = g & (NSORT - 1);
  if (b >= NBATCH) return;
  if (n >= NANCH) { keys[b*NSORT + n] = -INFINITY; idxs[b*NSORT + n] = n; return; }

  float imh = (float)ph[0], imw = (float)pw[0];
  float4 a = anchor_at(n);
  const float* d = deltas + ((size_t)b * NANCH + n) * 4;
  float aw = a.z - a.x, ah = a.w - a.y;
  float acx = a.x + 0.5f * aw, acy = a.y + 0.5f * ah;
  float cx = d[0] * aw + acx, cy = d[1] * ah + acy;
  float w  = aw * expf(d[2]), h = ah * expf(d[3]);
  float x1 = fminf(fmaxf(cx - 0.5f*w, 0.f), imw);
  float x2 = fminf(fmaxf(cx + 0.5f*w, 0.f), imw);
  float y1 = fminf(fmaxf(cy - 0.5f*h, 0.f), imh);
  float y2 = fminf(fmaxf(cy + 0.5f*h, 0.f), imh);
  float* bx = boxes + ((size_t)b * NANCH + n) * 4;
  bx[0] = x1; bx[1] = y1; bx[2] = x2; bx[3] = y2;
  float ws = x2 - x1, hs = y2 - y1;
  float sc = scores[(size_t)b * NANCH + n];
  keys[b*NSORT + n] = (ws > 16.0f && hs > 16.0f) ? sc : -INFINITY;
  idxs[b*NSORT + n] = n;
}

// ------------------------------------- per-batch bitonic sort in 256KB LDS
__global__ void sort_kernel(const float* __restrict__ keysG, const int* __restrict__ idxsG,
                            const float* __restrict__ boxes,
                            float* __restrict__ topv,      // [B][PRE_PAD]
                            float* __restrict__ bsorted) { // [B][PRE_PAD][4]
  extern __shared__ char smem[];
  float* k_ = (float*)smem;                       // 32768 * 4B
  int*   id_ = (int*)(smem + NSORT * sizeof(float));
  int b = blockIdx.x;
  for (int i = threadIdx.x; i < NSORT; i += blockDim.x) {
    k_[i] = keysG[b*NSORT + i];
    id_[i] = idxsG[b*NSORT + i];
  }
  __syncthreads();
  for (int k = 2; k <= NSORT; k <<= 1) {
    for (int j = k >> 1; j > 0; j >>= 1) {
      for (int i = threadIdx.x; i < NSORT; i += blockDim.x) {
        int ixj = i ^ j;
        if (ixj > i) {
          float ka = k_[i], kb = k_[ixj];
          int ia = id_[i], ib = id_[ixj];
          // "before" in final descending order: key desc, idx asc tie-break
          bool b_before_a = (kb > ka) || (kb == ka && ib < ia);
          bool a_before_b = (ka > kb) || (ka == kb && ia < ib);
          bool sw = ((i & k) == 0) ? b_before_a : a_before_b;
          if (sw) { k_[i] = kb; k_[ixj] = ka; id_[i] = ib; id_[ixj] = ia; }
        }
      }
      __syncthreads();
    }
  }
  for (int s = threadIdx.x; s < PRE_PAD; s += blockDim.x) {
    float kv = (s < PRE_NMS) ? k_[s] : -INFINITY;
    topv[b*PRE_PAD + s] = kv;
    float4 bx = make_float4(0.f, 0.f, 0.f, 0.f);
    if (s < PRE_NMS) {
      int id = id_[s];
      if (id < NANCH) bx = ((const float4*)boxes)[(size_t)b * NANCH + id];
    }
    ((float4*)bsorted)[(size_t)b * PRE_PAD + s] = bx;
  }
}

// ------------------- WMMA suppression-bitmask build: one wave per 16x32 tile
// areaSum[16][16] = area_i + area_j computed by V_WMMA_F32_16X16X4_F32 via
//   A rows = (area_i, 1, 0, 0), B cols = (1, area_j, 0, 0)^T  (exact in f32)
__global__ void __launch_bounds__(32) mask_kernel(const float* __restrict__ bsorted,
                                                  unsigned* __restrict__ mask) {
  int jw = blockIdx.x;              // 0..187 : 32 columns
  int it = blockIdx.y;              // 0..374 : 16 rows
  int b  = blockIdx.z;
  int lane = threadIdx.x;           // 0..31
  int i0 = it * 16;
  const float4* bs = (const float4*)bsorted + (size_t)b * PRE_PAD;

  __shared__ float4 bi[16];
  if (lane < 16) bi[lane] = bs[i0 + lane];
  __syncthreads();

  float4 bm0 = bi[lane & 15];
  float areaI = (bm0.z - bm0.x) * (bm0.w - bm0.y);

  // A matrix 16x4 (f32): lanes 0-15 hold {K0,K1} in {v0,v1}; lanes 16-31 hold {K2,K3}=0
  v2f Av;
  Av.x = (lane < 16) ? areaI : 0.f;   // K0 = area_i
  Av.y = (lane < 16) ? 1.0f  : 0.f;   // K1 = 1
  unsigned wbits = 0;

  #pragma unroll
  for (int half = 0; half < 2; ++half) {
    int j0 = jw * 32 + half * 16;
    float4 bj = bs[j0 + (lane & 15)];            // column box (replicated in hi lanes)
    float areaJ = (bj.z - bj.x) * (bj.w - bj.y);
    // B matrix 4x16 (f32): row K0 = 1, row K1 = area_j, K2/K3 = 0
    v2f Bv;
    Bv.x = (lane < 16) ? 1.0f  : 0.f;
    Bv.y = (lane < 16) ? areaJ : 0.f;
    v8f C = {};
    v8f D = __builtin_amdgcn_wmma_f32_16x16x4_f32(
        false, Av, false, Bv, (short)0, C, false, false);

    #pragma unroll
    for (int r = 0; r < 8; ++r) {
      int m = (lane < 16) ? r : r + 8;           // D layout: vgpr r, lo lanes M=r, hi lanes M=r+8
      float4 bm = bi[m];
      float ix1 = fmaxf(bm.x, bj.x), iy1 = fmaxf(bm.y, bj.y);
      float ix2 = fminf(bm.z, bj.z), iy2 = fminf(bm.w, bj.w);
      float inter = fmaxf(ix2 - ix1, 0.f) * fmaxf(iy2 - iy1, 0.f);
      float den = D[r] - inter + 1e-6f;          // area_i + area_j - inter + eps  (>0)
      bool bit = inter > NMS_THR * den;          // iou > 0.7
      unsigned bal = __builtin_amdgcn_ballot_w32(bit);
      int myrow = lane & 15;
      if (myrow == r)     wbits |= ((bal & 0xFFFFu) << (16 * half));
      if (myrow == r + 8) wbits |= (((bal >> 16) & 0xFFFFu) << (16 * half));
    }
  }
  if (lane < 16)
    mask[((size_t)b * PRE_NMS + (i0 + lane)) * NWORDS + jw] = wbits;
}

// ---------------------- greedy NMS scan over bitmask + output writes
__global__ void nms_scan_kernel(const float* __restrict__ topv,
                                const float* __restrict__ bsorted,
                                const unsigned* __restrict__ mask,
                                float* __restrict__ out) {
  int b = blockIdx.x;
  int tid = threadIdx.x;  // 256
  __shared__ unsigned sup[NWORDS];
  __shared__ int picks[POST_NMS];
  __shared__ int first;

  if (tid < NWORDS) {
    unsigned w = 0;
    for (int k = 0; k < 32; ++k) {
      int j = tid * 32 + k;
      bool suppressed = (j >= PRE_NMS) || (topv[b*PRE_PAD + j] == -INFINITY);
      if (suppressed) w |= (1u << k);
    }
    sup[tid] = w;
  }
  __syncthreads();

  for (int p = 0; p < POST_NMS; ++p) {
    if (tid == 0) first = 1 << 30;
    __syncthreads();
    if (tid < NWORDS) {
      unsigned w = sup[tid];
      if (w != 0xFFFFFFFFu) atomicMin(&first, tid * 32 + (__ffs(~w) - 1));
    }
    __syncthreads();
    int j = first;
    if (j < PRE_NMS) {
      if (tid < NWORDS) sup[tid] |= mask[((size_t)b * PRE_NMS + j) * NWORDS + tid];
    }
    if (tid == 0) picks[p] = (j < PRE_NMS) ? j : -1;
    __syncthreads();
  }

  for (int p = tid; p < POST_NMS; p += blockDim.x) {
    int j = picks[p];
    float4 bx = make_float4(0.f, 0.f, 0.f, 0.f);
    if (j >= 0) bx = ((const float4*)bsorted)[(size_t)b * PRE_PAD + j];
    int row = b * POST_NMS + p;
    out[row*4 + 0] = bx.x; out[row*4 + 1] = bx.y;
    out[row*4 + 2] = bx.z; out[row*4 + 3] = bx.w;
    out[4800 + row]  = (float)b;                 // rois_idx
    out[96000 + row] = (j >= 0) ? 1.0f : 0.0f;   // valid
  }
}

// ---------------------------------------------------------------- launch
extern "C" void kernel_launch(void* const* d_in, const int* in_sizes, int n_in,
                              void* d_out, int out_size, void* d_ws, size_t ws_size,
                              hipStream_t stream) {
  const float* deltas = (const float*)d_in[1];   // [B, N, 4]
  const float* scores = (const float*)d_in[2];   // [B, N]
  const int*   ph     = (const int*)d_in[3];
  const int*   pw     = (const int*)d_in[4];
  float* out = (float*)d_out;

  // workspace carve-up (16B aligned)
  char* ws = (char*)d_ws;
  size_t off = 0;
  float* boxes   = (float*)(ws + off); off += (size_t)NBATCH * NANCH * 4 * sizeof(float);   // 1.44 MB
  float* keys    = (float*)(ws + off); off += (size_t)NBATCH * NSORT * sizeof(float);       // 0.5 MB
  int*   idxs    = (int*)  (ws + off); off += (size_t)NBATCH * NSORT * sizeof(int);         // 0.5 MB
  float* topv    = (float*)(ws + off); off += (size_t)NBATCH * PRE_PAD * sizeof(float);
  float* bsorted = (float*)(ws + off); off += (size_t)NBATCH * PRE_PAD * 4 * sizeof(float);
  unsigned* mask = (unsigned*)(ws + off);        // 18 MB, L2-resident

  // 256KB dynamic LDS for the per-batch sort (CDNA5 allows 320KB/WG)
  hipFuncSetAttribute((const void*)sort_kernel,
                      hipFuncAttributeMaxDynamicSharedMemorySize,
                      NSORT * (sizeof(float) + sizeof(int)));

  anchors_kernel<<<(NANCH + 255) / 256, 256, 0, stream>>>(out);
  decode_kernel<<<(NBATCH * NSORT) / 256, 256, 0, stream>>>(
      deltas, scores, ph, pw, boxes, keys, idxs);
  sort_kernel<<<NBATCH, 1024, NSORT * (sizeof(float) + sizeof(int)), stream>>>(
      keys, idxs, boxes, topv, bsorted);
  mask_kernel<<<dim3(NWORDS, PRE_NMS / 16, NBATCH), 32, 0, stream>>>(bsorted, mask);
  nms_scan_kernel<<<NBATCH, 256, 0, stream>>>(topv, bsorted, mask, out);
}